// ToxGNN_56435870269948
// MI455X (gfx1250) — compile-verified
//
#include <hip/hip_runtime.h>
#include <hip/hip_bf16.h>
#include <math.h>

typedef float v2f __attribute__((ext_vector_type(2)));
typedef float v8f __attribute__((ext_vector_type(8)));
typedef unsigned int u32x4 __attribute__((ext_vector_type(4)));
typedef int i32x4 __attribute__((ext_vector_type(4)));
typedef int i32x8 __attribute__((ext_vector_type(8)));

#define H 256
#define G3 768
#define F_NODE 133
#define F_EDGE 7
#define GK 263      // H + F_EDGE
#define GKP 264     // padded
#define NGRAPH 4096
#define TOUT 12

#if defined(__AMDGCN__) && __has_builtin(__builtin_amdgcn_tensor_load_to_lds)
#define HAVE_TDM 1
#endif

// ---------------------------------------------------------------------------
// TDM 2D tile load: global[rows x cols slice] -> LDS, row stride = stride_elems,
// 4-byte elements, zero-fill beyond (rem_cols, rem_rows), LDS padding of
// 1 DWORD after every 32 DWORDs (pad_interval=4 -> 32 dwords, pad_amount=0 -> 1).
// Descriptor layout per CDNA5 ISA section 8.3/8.4.
// ---------------------------------------------------------------------------
__device__ __forceinline__ void tdm_load_2d(unsigned lds_off, const float* gptr,
                                            int rem_cols, int rem_rows,
                                            int tile_cols, int tile_rows,
                                            int stride_elems)
{
#ifdef HAVE_TDM
    unsigned long long ga = (unsigned long long)(size_t)gptr;
    unsigned rc = (rem_cols > 0) ? (unsigned)rem_cols : 0u;
    unsigned rr = (rem_rows > 0) ? (unsigned)rem_rows : 0u;

    u32x4 g0;
    g0[0] = 1u;                                             // count=1 (valid user D#)
    g0[1] = lds_off;                                        // lds_addr (bytes)
    g0[2] = (unsigned)(ga & 0xFFFFFFFFu);                   // global_addr[31:0]
    g0[3] = (unsigned)((ga >> 32) & 0x01FFFFFFu)            // global_addr[56:32]
            | 0x80000000u;                                  // type=2 ("image")

    i32x8 g1;
    g1[0] = (int)((2u << 16)        // data_size = 4 bytes
                | (1u << 20)        // pad_enable
                | (4u << 22));      // pad_interval: 32 DWORDs (pad_amount=0 -> 1 DWORD)
    g1[1] = (int)((rc & 0xFFFFu) << 16);                          // tensor_dim0[15:0]
    g1[2] = (int)(((rc >> 16) & 0xFFFFu) | ((rr & 0xFFFFu) << 16)); // dim0 hi | dim1 lo
    g1[3] = (int)(((rr >> 16) & 0xFFFFu) | ((unsigned)tile_cols << 16)); // dim1 hi | tile_dim0
    g1[4] = (int)((unsigned)tile_rows & 0xFFFFu);                 // tile_dim1 (tile_dim2=0)
    g1[5] = (int)stride_elems;                                    // tensor_dim0_stride[31:0]
    g1[6] = 0;                                                    // stride hi | dim1_stride lo
    g1[7] = 0;

    i32x4 z4 = (i32x4){0, 0, 0, 0};
#if __clang_major__ >= 23
    i32x8 z8 = (i32x8){0, 0, 0, 0, 0, 0, 0, 0};
    __builtin_amdgcn_tensor_load_to_lds(g0, g1, z4, z4, z8, 0);
#else
    __builtin_amdgcn_tensor_load_to_lds(g0, g1, z4, z4, 0);
#endif
#endif
}

// ---------------------------------------------------------------------------
// Generic WMMA f32 GEMM:  C[M,Nc] = out_act( A[M,K] @ W[Nc,K]^T + bias )
// a_act: 0=none 1=elu applied to A at fragment read.
// out_act: 0=none 1=lrelu 2=relu
// Operand staging via TDM (double-buffered) when available.
// ---------------------------------------------------------------------------
#define BM 128
#define BN 64
#define KC 32

__global__ __launch_bounds__(256)
void wmma_gemm_kernel(const float* __restrict__ A, const float* __restrict__ W,
                      const float* __restrict__ bias, float* __restrict__ C,
                      int M, int K, int Nc, int a_act, int out_act)
{
#ifdef HAVE_TDM
    __shared__ float As[2][BM][KC + 1];
    __shared__ float Ws[2][BN][KC + 1];
#else
    __shared__ float As[1][BM][KC + 1];
    __shared__ float Ws[1][BN][KC + 1];
#endif

    const int tid  = threadIdx.x;
    const int wave = tid >> 5;
    const int lane = tid & 31;
    const int hl   = lane & 15;
    const int m0   = blockIdx.y * BM;
    const int n0   = blockIdx.x * BN;
    const int nchunks = (K + KC - 1) / KC;

    v8f acc[4];
    #pragma unroll
    for (int t = 0; t < 4; ++t) acc[t] = (v8f){0.f,0.f,0.f,0.f,0.f,0.f,0.f,0.f};

#ifdef HAVE_TDM
    if (wave == 0) {
        tdm_load_2d((unsigned)(size_t)&As[0][0][0], A + (size_t)m0 * K,
                    K, M - m0, KC, BM, K);
        tdm_load_2d((unsigned)(size_t)&Ws[0][0][0], W + (size_t)n0 * K,
                    K, Nc - n0, KC, BN, K);
    }
#endif

    for (int c = 0; c < nchunks; ++c) {
        const int k0 = c * KC;
#ifdef HAVE_TDM
        const int cur = c & 1;
        if (wave == 0) {
            if (c + 1 < nchunks) {
                const int k1 = k0 + KC;
                tdm_load_2d((unsigned)(size_t)&As[cur ^ 1][0][0],
                            A + (size_t)m0 * K + k1, K - k1, M - m0, KC, BM, K);
                tdm_load_2d((unsigned)(size_t)&Ws[cur ^ 1][0][0],
                            W + (size_t)n0 * K + k1, K - k1, Nc - n0, KC, BN, K);
                __builtin_amdgcn_s_wait_tensorcnt((short)2);  // pair c complete (in-order)
            } else {
                __builtin_amdgcn_s_wait_tensorcnt((short)0);
            }
        }
        __syncthreads();
#else
        const int cur = 0;
        #pragma unroll
        for (int i = 0; i < (BM * KC) / 256; ++i) {
            int e = tid + i * 256;
            int r = e >> 5, kk = e & 31;
            int gr = m0 + r, gk = k0 + kk;
            float v = 0.f;
            if (gr < M && gk < K) v = A[(size_t)gr * K + gk];
            As[0][r][kk] = v;
        }
        #pragma unroll
        for (int i = 0; i < (BN * KC) / 256; ++i) {
            int e = tid + i * 256;
            int r = e >> 5, kk = e & 31;
            int gn = n0 + r, gk = k0 + kk;
            float v = 0.f;
            if (gn < Nc && gk < K) v = W[(size_t)gn * K + gk];
            Ws[0][r][kk] = v;
        }
        __syncthreads();
#endif

        const int rowA = wave * 16 + hl;
        const int koff = (lane < 16) ? 0 : 2;
        #pragma unroll
        for (int kk = 0; kk < KC; kk += 4) {
            v2f a;
            a.x = As[cur][rowA][kk + koff];
            a.y = As[cur][rowA][kk + koff + 1];
            if (a_act == 1) {
                a.x = (a.x > 0.f) ? a.x : (expf(a.x) - 1.f);
                a.y = (a.y > 0.f) ? a.y : (expf(a.y) - 1.f);
            }
            #pragma unroll
            for (int t = 0; t < 4; ++t) {
                int nn = t * 16 + hl;
                v2f b;
                b.x = Ws[cur][nn][kk + koff];
                b.y = Ws[cur][nn][kk + koff + 1];
                acc[t] = __builtin_amdgcn_wmma_f32_16x16x4_f32(
                    false, a, false, b, (short)0, acc[t], false, false);
            }
        }
        __syncthreads();
    }

    // C layout: VGPR i, lanes 0-15 -> M=i, lanes 16-31 -> M=8+i
    const int rbase = m0 + wave * 16 + ((lane < 16) ? 0 : 8);
    #pragma unroll
    for (int t = 0; t < 4; ++t) {
        int col = n0 + t * 16 + hl;
        float bv = (bias != nullptr && col < Nc) ? bias[col] : 0.f;
        #pragma unroll
        for (int i = 0; i < 8; ++i) {
            int row = rbase + i;
            if (row < M && col < Nc) {
                float v = acc[t][i] + bv;
                if (out_act == 1)      v = (v > 0.f) ? v : 0.01f * v;
                else if (out_act == 2) v = (v > 0.f) ? v : 0.f;
                C[(size_t)row * Nc + col] = v;
            }
        }
    }
}

// ---------------------------------------------------------------------------
// Fused GATEConv edge-logit kernel (WMMA):
//   alpha[e] = lrelu( sum_n lrelu( (W1 @ cat(x0[src],ea))[n] ) * att_l[n] + rdst[dst[e]] )
// 32 edges per block, K=263 (padded to 264), avoids materializing [E,256] xj.
// ---------------------------------------------------------------------------
#define TE 32

__global__ __launch_bounds__(256)
void gate_edge_logit_kernel(const float* __restrict__ x0, const float* __restrict__ ea,
                            const float* __restrict__ W1, const float* __restrict__ att_l,
                            const float* __restrict__ rdst, const int* __restrict__ src,
                            const int* __restrict__ dst, float* __restrict__ alpha, int E_)
{
    __shared__ float Xe[TE][GKP];
    __shared__ float partial[TE];

    const int tid = threadIdx.x;
    const int e0  = blockIdx.x * TE;

    for (int idx = tid; idx < TE * GKP; idx += 256) {
        int r = idx / GKP, c = idx - r * GKP;
        int e = e0 + r;
        float v = 0.f;
        if (e < E_) {
            if (c < H)        v = x0[(size_t)src[e] * H + c];
            else if (c < GK)  v = ea[(size_t)e * F_EDGE + (c - H)];
        }
        Xe[r][c] = v;
    }
    if (tid < TE) partial[tid] = 0.f;
    __syncthreads();

    const int wave = tid >> 5, lane = tid & 31, hl = lane & 15;
    const int mi     = wave & 1;          // 2 row-tiles of 16
    const int ntBase = (wave >> 1) * 4;   // 4 col-tiles of 16 per wave
    const int rowA   = mi * 16 + hl;
    const int koff   = (lane < 16) ? 0 : 2;

    v8f acc[4];
    #pragma unroll
    for (int t = 0; t < 4; ++t) acc[t] = (v8f){0.f,0.f,0.f,0.f,0.f,0.f,0.f,0.f};

    for (int kk = 0; kk < GKP; kk += 4) {
        v2f a;
        a.x = Xe[rowA][kk + koff];
        a.y = Xe[rowA][kk + koff + 1];
        int kb = kk + koff;
        #pragma unroll
        for (int t = 0; t < 4; ++t) {
            int n = (ntBase + t) * 16 + hl;
            v2f b;
            b.x = (kb < GK)     ? W1[(size_t)n * GK + kb]     : 0.f;
            b.y = (kb + 1 < GK) ? W1[(size_t)n * GK + kb + 1] : 0.f;
            acc[t] = __builtin_amdgcn_wmma_f32_16x16x4_f32(
                false, a, false, b, (short)0, acc[t], false, false);
        }
    }

    // lrelu, weight by att_l, reduce over columns per edge row
    float rsum[8];
    #pragma unroll
    for (int i = 0; i < 8; ++i) rsum[i] = 0.f;
    #pragma unroll
    for (int t = 0; t < 4; ++t) {
        int col = (ntBase + t) * 16 + hl;
        float w = att_l[col];
        #pragma unroll
        for (int i = 0; i < 8; ++i) {
            float v = acc[t][i];
            v = (v > 0.f) ? v : 0.01f * v;
            rsum[i] += v * w;
        }
    }
    int rb = mi * 16 + ((lane < 16) ? 0 : 8);
    #pragma unroll
    for (int i = 0; i < 8; ++i) atomicAdd(&partial[rb + i], rsum[i]);
    __syncthreads();

    if (tid < TE) {
        int e = e0 + tid;
        if (e < E_) {
            float v = partial[tid] + rdst[dst[e]];
            alpha[e] = (v > 0.f) ? v : 0.01f * v;
        }
    }
}

// ---------------------------------------------------------------------------
// Elementwise / scatter / reduction helpers
// ---------------------------------------------------------------------------
__global__ void rowdot_kernel(const float* __restrict__ A, const float* __restrict__ v,
                              float* __restrict__ out, int M, int K)
{
    int w = blockIdx.x * 8 + (threadIdx.x >> 5);
    int lane = threadIdx.x & 31;
    if (w >= M) return;
    const float* row = A + (size_t)w * K;
    float s = 0.f;
    for (int k = lane; k < K; k += 32) s += row[k] * v[k];
    #pragma unroll
    for (int off = 16; off > 0; off >>= 1) s += __shfl_xor(s, off, 32);
    if (lane == 0) out[w] = s;
}

__global__ void fill_val_kernel(float* __restrict__ p, float v, long n) {
    long i = (long)blockIdx.x * 256 + threadIdx.x;
    if (i < n) p[i] = v;
}

__global__ void fill_bias_kernel(float* __restrict__ p, const float* __restrict__ bias, long rows) {
    long i = (long)blockIdx.x * 256 + threadIdx.x;
    if (i < rows * H) p[i] = bias[i & (H - 1)];
}

__global__ void relu_kernel(float* __restrict__ p, long n) {
    long i = (long)blockIdx.x * 256 + threadIdx.x;
    if (i < n) { float v = p[i]; p[i] = (v > 0.f) ? v : 0.f; }
}

__global__ void edge_logit_kernel(const float* __restrict__ rA, const float* __restrict__ rB,
                                  const int* __restrict__ src, const int* __restrict__ dst,
                                  float* __restrict__ alpha, int n)
{
    int e = blockIdx.x * 256 + threadIdx.x;
    if (e >= n) return;
    float v = rA[src[e]] + rB[dst[e]];
    alpha[e] = (v > 0.f) ? v : 0.01f * v;
}

__global__ void mol_logit_kernel(const float* __restrict__ a_src, const float* __restrict__ rG,
                                 const int* __restrict__ batch, float* __restrict__ alpha, int n)
{
    int i = blockIdx.x * 256 + threadIdx.x;
    if (i >= n) return;
    float v = a_src[i] + rG[batch[i]];
    alpha[i] = (v > 0.f) ? v : 0.01f * v;
}

__global__ void seg_max_kernel(const float* __restrict__ logits, const int* __restrict__ seg,
                               float* __restrict__ mx, int n)
{
    int i = blockIdx.x * 256 + threadIdx.x;
    if (i >= n) return;
    float v = logits[i];
    float* addr = &mx[seg[i]];
    if (v >= 0.f) atomicMax((int*)addr, __float_as_int(v));
    else          atomicMin((unsigned int*)addr, __float_as_uint(v));
}

__global__ void seg_exp_sum_kernel(float* __restrict__ logits, const int* __restrict__ seg,
                                   const float* __restrict__ mx, float* __restrict__ sum, int n)
{
    int i = blockIdx.x * 256 + threadIdx.x;
    if (i >= n) return;
    int s = seg[i];
    float v = expf(logits[i] - mx[s]);
    logits[i] = v;
    atomicAdd(&sum[s], v);
}

__global__ void seg_norm_kernel(float* __restrict__ logits, const int* __restrict__ seg,
                                const float* __restrict__ sum, int n)
{
    int i = blockIdx.x * 256 + threadIdx.x;
    if (i >= n) return;
    logits[i] = logits[i] / (sum[seg[i]] + 1e-16f);
}

// one block per edge, 256 lanes = 256 channels
__global__ __launch_bounds__(256)
void scatter_msg_kernel(const float* __restrict__ xsrc, const float* __restrict__ alpha,
                        const int* __restrict__ src, const int* __restrict__ dst,
                        float* __restrict__ hout, int E_)
{
    int e = blockIdx.x;
    if (e >= E_) return;
    int s = src[e], d = dst[e];
    float a = alpha[e];
    int c = threadIdx.x;
    atomicAdd(&hout[(size_t)d * H + c], xsrc[(size_t)s * H + c] * a);
}

// one block per node; alpha==nullptr -> weight 1
__global__ __launch_bounds__(256)
void scatter_node_kernel(const float* __restrict__ xs, const float* __restrict__ alpha,
                         const int* __restrict__ seg, float* __restrict__ out, int M)
{
    int i = blockIdx.x;
    if (i >= M) return;
    int g = seg[i];
    float a = (alpha != nullptr) ? alpha[i] : 1.f;
    int c = threadIdx.x;
    atomicAdd(&out[(size_t)g * H + c], xs[(size_t)i * H + c] * a);
}

// GRU gate combine (torch GRUCell, gate order r,z,n); optional ReLU on output
__global__ void gru_gate_kernel(const float* __restrict__ gi, const float* __restrict__ gh,
                                const float* __restrict__ hprev, float* __restrict__ out,
                                long n, int relu_out)
{
    long idx = (long)blockIdx.x * 256 + threadIdx.x;
    if (idx >= n) return;
    long row = idx >> 8;
    int c = (int)(idx & 255);
    const float* gir = gi + row * G3;
    const float* ghr = gh + row * G3;
    float i_r = gir[c], i_z = gir[H + c], i_n = gir[2 * H + c];
    float h_r = ghr[c], h_z = ghr[H + c], h_n = ghr[2 * H + c];
    float r = 1.f / (1.f + expf(-(i_r + h_r)));
    float z = 1.f / (1.f + expf(-(i_z + h_z)));
    float nn = tanhf(i_n + r * h_n);
    float o = (1.f - z) * nn + z * hprev[idx];
    if (relu_out) o = (o > 0.f) ? o : 0.f;
    out[idx] = o;
}

// final fc: out[g,t] = emb[g]·fc_w[t] + fc_b[t]; one wave per output
__global__ __launch_bounds__(256)
void fc_kernel(const float* __restrict__ emb, const float* __restrict__ fw,
               const float* __restrict__ fb, float* __restrict__ out, int Gn)
{
    int w = blockIdx.x * 8 + (threadIdx.x >> 5);
    int lane = threadIdx.x & 31;
    if (w >= Gn * TOUT) return;
    int g = w / TOUT, t = w - g * TOUT;
    const float* er = emb + (size_t)g * H;
    const float* wr = fw + (size_t)t * H;
    float s = 0.f;
    for (int k = lane; k < H; k += 32) s += er[k] * wr[k];
    #pragma unroll
    for (int off = 16; off > 0; off >>= 1) s += __shfl_xor(s, off, 32);
    if (lane == 0) out[w] = s + fb[t];
}

// ---------------------------------------------------------------------------
static inline int ceil_div(long a, long b) { return (int)((a + b - 1) / b); }

extern "C" void kernel_launch(void* const* d_in, const int* in_sizes, int n_in,
                              void* d_out, int out_size, void* d_ws, size_t ws_size,
                              hipStream_t stream)
{
    const float* x           = (const float*)d_in[0];
    const float* edge_attr   = (const float*)d_in[1];
    const float* lin1_w      = (const float*)d_in[2];
    const float* lin1_b      = (const float*)d_in[3];
    const float* gate_att_l  = (const float*)d_in[4];
    const float* gate_att_r  = (const float*)d_in[5];
    const float* gate_lin1_w = (const float*)d_in[6];
    const float* gate_lin2_w = (const float*)d_in[7];
    const float* gate_bias   = (const float*)d_in[8];
    const float* atom_lin_w  = (const float*)d_in[9];
    const float* atom_att_s  = (const float*)d_in[10];
    const float* atom_att_d  = (const float*)d_in[11];
    const float* atom_bias   = (const float*)d_in[12];
    const float* gru_w_ih    = (const float*)d_in[13];
    const float* gru_w_hh    = (const float*)d_in[14];
    const float* gru_b_ih    = (const float*)d_in[15];
    const float* gru_b_hh    = (const float*)d_in[16];
    const float* mol_lin_w   = (const float*)d_in[17];
    const float* mol_att_s   = (const float*)d_in[18];
    const float* mol_att_d   = (const float*)d_in[19];
    const float* mol_bias    = (const float*)d_in[20];
    const float* mg_w_ih     = (const float*)d_in[21];
    const float* mg_w_hh     = (const float*)d_in[22];
    const float* mg_b_ih     = (const float*)d_in[23];
    const float* mg_b_hh     = (const float*)d_in[24];
    const float* lin2_w      = (const float*)d_in[25];
    const float* lin2_b      = (const float*)d_in[26];
    const float* fc_w        = (const float*)d_in[27];
    const float* fc_b        = (const float*)d_in[28];
    const int*   edge_index  = (const int*)d_in[29];
    const int*   batch       = (const int*)d_in[30];

    const int N = in_sizes[30];
    const int E = in_sizes[29] / 2;
    const int Gn = NGRAPH;
    const int* src = edge_index;
    const int* dst = edge_index + E;

    // workspace carve-up
    float* ws = (float*)d_ws;
    size_t off = 0;
    auto alloc = [&](size_t n) { float* p = ws + off; off += n; return p; };
    float* x0    = alloc((size_t)N * H);
    float* xcur  = alloc((size_t)N * H);
    float* xtmp  = alloc((size_t)N * H);
    float* hbuf  = alloc((size_t)N * H);
    float* gi    = alloc((size_t)N * G3);
    float* gh    = alloc((size_t)N * G3);
    float* rA    = alloc((size_t)N);
    float* rB    = alloc((size_t)N);
    float* alpha = alloc((size_t)E);
    float* smax  = alloc((size_t)N);
    float* ssum  = alloc((size_t)N);
    float* outg  = alloc((size_t)Gn * H);
    float* gtmp  = alloc((size_t)Gn * H);
    float* hg    = alloc((size_t)Gn * H);
    float* gig   = alloc((size_t)Gn * G3);
    float* ghg   = alloc((size_t)Gn * G3);
    float* emb   = alloc((size_t)Gn * H);

    auto gemm = [&](const float* A, const float* W, const float* b, float* C,
                    int M, int K, int Nc, int a_act, int o_act) {
        dim3 grid(ceil_div(Nc, BN), ceil_div(M, BM));
        wmma_gemm_kernel<<<grid, 256, 0, stream>>>(A, W, b, C, M, K, Nc, a_act, o_act);
    };
    auto seg_softmax = [&](float* logits, const int* seg, int n_el, int n_seg) {
        fill_val_kernel<<<ceil_div(n_seg, 256), 256, 0, stream>>>(smax, -3.4e38f, n_seg);
        fill_val_kernel<<<ceil_div(n_seg, 256), 256, 0, stream>>>(ssum, 0.f, n_seg);
        seg_max_kernel<<<ceil_div(n_el, 256), 256, 0, stream>>>(logits, seg, smax, n_el);
        seg_exp_sum_kernel<<<ceil_div(n_el, 256), 256, 0, stream>>>(logits, seg, smax, ssum, n_el);
        seg_norm_kernel<<<ceil_div(n_el, 256), 256, 0, stream>>>(logits, seg, ssum, n_el);
    };

    // ---- x0 = lrelu(x @ lin1_w^T + b) ----
    gemm(x, lin1_w, lin1_b, x0, N, F_NODE, H, 0, 1);

    // ---- GATEConv ----
    rowdot_kernel<<<ceil_div(N, 8), 256, 0, stream>>>(x0, gate_att_r, rB, N, H);
    gate_edge_logit_kernel<<<ceil_div(E, TE), 256, 0, stream>>>(
        x0, edge_attr, gate_lin1_w, gate_att_l, rB, src, dst, alpha, E);
    seg_softmax(alpha, dst, E, N);
    gemm(x0, gate_lin2_w, nullptr, xtmp, N, H, H, 0, 0);
    fill_bias_kernel<<<ceil_div((long)N * H, 256), 256, 0, stream>>>(hbuf, gate_bias, N);
    scatter_msg_kernel<<<E, 256, 0, stream>>>(xtmp, alpha, src, dst, hbuf, E);
    gemm(hbuf, gru_w_ih, gru_b_ih, gi, N, H, G3, 1, 0);   // elu(h) @ w_ih^T
    gemm(x0,   gru_w_hh, gru_b_hh, gh, N, H, G3, 0, 0);
    gru_gate_kernel<<<ceil_div((long)N * H, 256), 256, 0, stream>>>(
        gi, gh, x0, xcur, (long)N * H, 1);

    // ---- 3 GATConv layers ----
    for (int l = 0; l < 3; ++l) {
        gemm(xcur, atom_lin_w + (size_t)l * H * H, nullptr, xtmp, N, H, H, 0, 0);
        rowdot_kernel<<<ceil_div(N, 8), 256, 0, stream>>>(xtmp, atom_att_s + l * H, rA, N, H);
        rowdot_kernel<<<ceil_div(N, 8), 256, 0, stream>>>(xtmp, atom_att_d + l * H, rB, N, H);
        edge_logit_kernel<<<ceil_div(E, 256), 256, 0, stream>>>(rA, rB, src, dst, alpha, E);
        seg_softmax(alpha, dst, E, N);
        fill_bias_kernel<<<ceil_div((long)N * H, 256), 256, 0, stream>>>(hbuf, atom_bias + l * H, N);
        scatter_msg_kernel<<<E, 256, 0, stream>>>(xtmp, alpha, src, dst, hbuf, E);
        gemm(hbuf, gru_w_ih + (size_t)(l + 1) * G3 * H, gru_b_ih + (l + 1) * G3, gi, N, H, G3, 1, 0);
        gemm(xcur, gru_w_hh + (size_t)(l + 1) * G3 * H, gru_b_hh + (l + 1) * G3, gh, N, H, G3, 0, 0);
        gru_gate_kernel<<<ceil_div((long)N * H, 256), 256, 0, stream>>>(
            gi, gh, xcur, xcur, (long)N * H, 1);
    }

    // ---- molecule pooling ----
    fill_val_kernel<<<ceil_div((long)Gn * H, 256), 256, 0, stream>>>(outg, 0.f, (long)Gn * H);
    scatter_node_kernel<<<N, 256, 0, stream>>>(xcur, nullptr, batch, outg, N);
    relu_kernel<<<ceil_div((long)Gn * H, 256), 256, 0, stream>>>(outg, (long)Gn * H);
    gemm(xcur, mol_lin_w, nullptr, xtmp, N, H, H, 0, 0);          // xm
    rowdot_kernel<<<ceil_div(N, 8), 256, 0, stream>>>(xtmp, mol_att_s, rA, N, H);

    for (int t = 0; t < 2; ++t) {
        gemm(outg, mol_lin_w, nullptr, gtmp, Gn, H, H, 0, 0);
        rowdot_kernel<<<ceil_div(Gn, 8), 256, 0, stream>>>(gtmp, mol_att_d, rB, Gn, H);
        mol_logit_kernel<<<ceil_div(N, 256), 256, 0, stream>>>(rA, rB, batch, alpha, N);
        seg_softmax(alpha, batch, N, Gn);
        fill_bias_kernel<<<ceil_div((long)Gn * H, 256), 256, 0, stream>>>(hg, mol_bias, Gn);
        scatter_node_kernel<<<N, 256, 0, stream>>>(xtmp, alpha, batch, hg, N);
        gemm(hg,   mg_w_ih, mg_b_ih, gig, Gn, H, G3, 1, 0);
        gemm(outg, mg_w_hh, mg_b_hh, ghg, Gn, H, G3, 0, 0);
        gru_gate_kernel<<<ceil_div((long)Gn * H, 256), 256, 0, stream>>>(
            gig, ghg, outg, outg, (long)Gn * H, 1);
    }

    // ---- heads ----
    gemm(outg, lin2_w, lin2_b, emb, Gn, H, H, 0, 0);
    fc_kernel<<<ceil_div((long)Gn * TOUT, 8), 256, 0, stream>>>(
        emb, fc_w, fc_b, (float*)d_out, Gn);
}